// ODE_23390391894829
// MI455X (gfx1250) — compile-verified
//
#include <hip/hip_runtime.h>
#include <hip/hip_bf16.h>

// Rod RK4 integrator for MI455X (gfx1250, wave32).
// State y[12] = {r0,r1,r2, R00..R22 (row-major)}, ux/uy constant per segment.
// Store-bound kernel (~381 MB out @ 23.3 TB/s): stage rows in LDS and emit
// coalesced non-temporal b128 stores.

#define TPB 256

typedef float v4f __attribute__((ext_vector_type(4)));
typedef float v2f __attribute__((ext_vector_type(2)));

__device__ __forceinline__ void ode_f(const float* __restrict__ y,
                                      float ux, float uy,
                                      float* __restrict__ k) {
  // dr = R[:,2]
  k[0] = y[5];  k[1] = y[8];  k[2] = y[11];
  // dR[i][0] = -R[i][2]*uy ; dR[i][1] = R[i][2]*ux ; dR[i][2] = R[i][0]*uy - R[i][1]*ux
#pragma unroll
  for (int i = 0; i < 3; ++i) {
    float Ri0 = y[3 + 3*i], Ri1 = y[4 + 3*i], Ri2 = y[5 + 3*i];
    k[3 + 3*i] = -Ri2 * uy;
    k[4 + 3*i] =  Ri2 * ux;
    k[5 + 3*i] = fmaf(Ri0, uy, -(Ri1 * ux));
  }
}

__device__ __forceinline__ void rk4_step(float* __restrict__ y, float ux, float uy) {
  const float h  = 0.005f;
  const float hh = (float)(0.5 * 0.005);       // 0.5*h as in reference (double -> f32)
  const float h6 = (float)(0.005 / 6.0);       // h/6  as in reference (double -> f32)
  float k1[12], k2[12], k3[12], k4[12], yt[12];
  ode_f(y, ux, uy, k1);
#pragma unroll
  for (int i = 0; i < 12; ++i) yt[i] = fmaf(hh, k1[i], y[i]);
  ode_f(yt, ux, uy, k2);
#pragma unroll
  for (int i = 0; i < 12; ++i) yt[i] = fmaf(hh, k2[i], y[i]);
  ode_f(yt, ux, uy, k3);
#pragma unroll
  for (int i = 0; i < 12; ++i) yt[i] = fmaf(h, k3[i], y[i]);
  ode_f(yt, ux, uy, k4);
#pragma unroll
  for (int i = 0; i < 12; ++i) {
    float s = ((k1[i] + 2.0f * k2[i]) + 2.0f * k3[i]) + k4[i];
    y[i] = fmaf(h6, s, y[i]);
  }
}

__global__ void __launch_bounds__(TPB)
rod_rk4_kernel(const float* __restrict__ act, float* __restrict__ out,
               int T, int batch) {
  __shared__ __align__(16) float lds[2][TPB * 14];

  const int tid = threadIdx.x;
  const int b   = blockIdx.x * TPB + tid;

  // gfx1250 prefetch path for the input vector
  __builtin_prefetch(act + (size_t)b * 6, 0, 3);

  const float a0 = act[(size_t)b * 6 + 0];
  const float a1 = act[(size_t)b * 6 + 1];
  const float a2 = act[(size_t)b * 6 + 2];
  // NOTE: reference's loop updates (l,ux,uy) with segment n's slice *after*
  // iteration n, so BOTH segments integrate with the n=0 action slice.
  // Columns 3..5 never affect the output.

  const float l  = 0.05f + a0;           // L0 + a[:,0]
  const float ld = l * 0.0075f;          // l * D
  const float ux = a2 / (-ld);           // a[:,2] / -(l*D)
  const float uy = a1 / ld;              // a[:,1] /  (l*D)
  int len = (int)floorf(l / 0.005f);     // floor(l / DS)
  if (len > T - 1) len = T - 1;

  float y[12] = {0.f, 0.f, 0.f,          // r
                 1.f, 0.f, 0.f,          // R = I
                 0.f, 1.f, 0.f,
                 0.f, 0.f, 1.f};

  int buf = 0;
#pragma unroll
  for (int seg = 0; seg < 2; ++seg) {
    for (int t = 0; t < T; ++t) {
      // sol_masked[t] = sol[min(t, len)] == stop stepping after `len` steps
      if (t > 0 && t <= len) rk4_step(y, ux, uy);

      // stage this row (14 floats per element) into LDS
      float* st = &lds[buf][tid * 14];
#pragma unroll
      for (int i = 0; i < 12; ++i) st[i] = y[i];
      st[12] = ux;
      st[13] = uy;
      __syncthreads();

      // block-contiguous region: 256 elems * 14 f32 = 14336 B, 16B aligned
      size_t G = ((size_t)(seg * T + t) * (size_t)batch +
                  (size_t)blockIdx.x * TPB) * 14;
      const v4f* s4 = (const v4f*)&lds[buf][0];
      v4f*       o4 = (v4f*)(out + G);
      __builtin_nontemporal_store(s4[tid],            o4 + tid);
      __builtin_nontemporal_store(s4[TPB + tid],      o4 + TPB + tid);
      __builtin_nontemporal_store(s4[2 * TPB + tid],  o4 + 2 * TPB + tid);
      const v2f* s2 = (const v2f*)&lds[buf][0];
      v2f*       o2 = (v2f*)(out + G);
      __builtin_nontemporal_store(s2[6 * TPB + tid],  o2 + 6 * TPB + tid);

      buf ^= 1;  // double buffer: next staging targets the other half
    }
  }
}

extern "C" void kernel_launch(void* const* d_in, const int* in_sizes, int n_in,
                              void* d_out, int out_size, void* d_ws, size_t ws_size,
                              hipStream_t stream) {
  (void)n_in; (void)d_ws; (void)ws_size;
  const float* act = (const float*)d_in[0];
  float*       out = (float*)d_out;

  const int batch = in_sizes[0] / 6;              // (BATCH, 6) actions
  const int T     = out_size / (2 * batch * 14);  // output is (2T, BATCH, 14)

  dim3 grid(batch / TPB), block(TPB);
  rod_rk4_kernel<<<grid, block, 0, stream>>>(act, out, T, batch);
}